// FrequencyFusion_28948079575615
// MI455X (gfx1250) — compile-verified
//
#include <hip/hip_runtime.h>
#include <hip/hip_bf16.h>
#include <math.h>

typedef __attribute__((ext_vector_type(2))) float v2f;
typedef __attribute__((ext_vector_type(8))) float v8f;

#define B_    32
#define L_    4096
#define C_    64
#define T_    1025      // STFT frame count (center=True)
#define F_    9         // n_fft/2+1
#define CH_   576       // C_*F_
#define TO_   1023      // conv output length (T_-2)
#define OF_   512       // out features
#define KK_   1728      // CH_*3
#define WIN_  128       // adaptive pool output

#define PI_F 3.14159265358979323846f

// ---------------------------------------------------------------------------
// Kernel 1: STFT magnitudes.  S[b][t][c*9+f] = |rDFT16(hann * frame)|_f
// Frames come from reflect-padded x (pad 8 each side), hop 4.
// ---------------------------------------------------------------------------
__global__ __launch_bounds__(256) void stft_mag_kernel(const float* __restrict__ x,
                                                       float* __restrict__ S) {
  __shared__ float wc[144];  // win[n]*cos(2pi f n/16), f in [0,9), n in [0,16)
  __shared__ float wsn[144]; // win[n]*sin(2pi f n/16)
  int tid = threadIdx.x;
  if (tid < 144) {
    int f = tid >> 4, n = tid & 15;
    float win = 0.5f * (1.0f - cosf(2.0f * PI_F * (float)n / 16.0f));
    float ang = 2.0f * PI_F * (float)(f * n) / 16.0f;
    wc[tid]  = win * cosf(ang);
    wsn[tid] = win * sinf(ang);
  }
  __syncthreads();

  long long g = (long long)blockIdx.x * 256 + tid;   // g = ((b*T + t)*C + c)
  int c = (int)(g & 63);
  int t = (int)((g >> 6) % T_);
  int b = (int)(g / ((long long)T_ * C_));
  if (b >= B_) return;

  float v[16];
#pragma unroll
  for (int n = 0; n < 16; ++n) {
    int j = t * 4 + n - 8;                 // index into reflect-padded signal
    if (j < 0) j = -j;
    else if (j >= L_) j = 2 * L_ - 2 - j;  // 'reflect' (no edge repeat)
    v[n] = x[((long long)b * L_ + j) * C_ + c];
  }

  float* srow = S + ((long long)b * T_ + t) * CH_ + c * F_;
#pragma unroll
  for (int f = 0; f < F_; ++f) {
    float re = 0.0f, im = 0.0f;
#pragma unroll
    for (int n = 0; n < 16; ++n) {
      re = fmaf(v[n], wc[f * 16 + n], re);
      im = fmaf(v[n], wsn[f * 16 + n], im);
    }
    srow[f] = sqrtf(re * re + im * im);
  }
}

// ---------------------------------------------------------------------------
// Kernel 2: reorder weight OIH [512][576][3] -> W2[k*576 + i][512]
// so that patch row (rows t,t+1,t+2 of S concatenated) matches K ordering.
// ---------------------------------------------------------------------------
__global__ __launch_bounds__(256) void wreorder_kernel(const float* __restrict__ w,
                                                       float* __restrict__ w2) {
  int g = blockIdx.x * 256 + threadIdx.x;
  if (g >= KK_ * OF_) return;
  int o  = g & (OF_ - 1);
  int kk = g >> 9;
  int k  = kk / CH_;
  int i  = kk - k * CH_;
  w2[g] = w[((long long)o * CH_ + i) * 3 + k];
}

// ---------------------------------------------------------------------------
// Kernel 3: implicit-GEMM conv via V_WMMA_F32_16X16X4_F32.
// Y[b][t][o] = bias[o] + sum_{kk<1728} S[b][t + kk/576][kk%576] * W2[kk][o]
// Workgroup: 256 threads (8 waves), tile 128(t) x 128(o); wave = 32x64.
// K staged through LDS in chunks of 32 (576 % 32 == 0 -> no tap crossing).
// ---------------------------------------------------------------------------
__global__ __launch_bounds__(256) void conv_gemm_kernel(const float* __restrict__ S,
                                                        const float* __restrict__ W2,
                                                        const float* __restrict__ bias,
                                                        float* __restrict__ Y) {
  __shared__ float lA[128 * 36];   // 128 t-rows x 32 K, stride 36 (16B aligned, no bank conflict)
  __shared__ float lB[32 * 132];   // 32 K-rows x 128 o, stride 132

  int gid  = blockIdx.x;       // b*32 + tT*4 + tO
  int b    = gid >> 5;
  int rem  = gid & 31;
  int tT   = rem >> 2;         // 0..7  (t tiles of 128; last one partial)
  int tO   = rem & 3;          // 0..3  (o tiles of 128)
  int tBase = tT * 128;
  int oBase = tO * 128;

  int tid   = threadIdx.x;
  int lane  = tid & 31;
  int wave  = tid >> 5;        // 0..7
  int waveT = wave & 3;        // 4 waves along t (32 rows each)
  int waveO = wave >> 2;       // 2 waves along o (64 cols each)
  int half  = lane >> 4;       // A/B fragment K-half select
  int ln    = lane & 15;       // M / N within fragment
  int k0    = half * 2;

  v8f acc[2][4];
#pragma unroll
  for (int ts = 0; ts < 2; ++ts)
#pragma unroll
    for (int os = 0; os < 4; ++os)
      acc[ts][os] = (v8f){0.f, 0.f, 0.f, 0.f, 0.f, 0.f, 0.f, 0.f};

  const float* Sb = S + (long long)b * T_ * CH_;

  for (int kb = 0; kb < KK_; kb += 32) {
    int r    = kb / CH_;          // conv tap 0..2 (constant within chunk)
    int col0 = kb - r * CH_;      // column base within S row (multiple of 32)

    // Stage A: 128 rows x 32 floats = 1024 float4, 4 per thread.
#pragma unroll
    for (int it = 0; it < 4; ++it) {
      int idx = tid + it * 256;
      int row = idx >> 3;
      int c4  = idx & 7;
      int gr  = tBase + row + r;
      gr = gr > (T_ - 1) ? (T_ - 1) : gr;   // clamp (results for t>=1023 discarded)
      float4 val = *(const float4*)(Sb + (long long)gr * CH_ + col0 + c4 * 4);
      *(float4*)(&lA[row * 36 + c4 * 4]) = val;
    }
    // Stage B: 32 rows x 128 floats = 1024 float4.
#pragma unroll
    for (int it = 0; it < 4; ++it) {
      int idx = tid + it * 256;
      int kl  = idx >> 5;
      int c4  = idx & 31;
      float4 val = *(const float4*)(W2 + (long long)(kb + kl) * OF_ + oBase + c4 * 4);
      *(float4*)(&lB[kl * 132 + c4 * 4]) = val;
    }
    __syncthreads();

#pragma unroll
    for (int kk0 = 0; kk0 < 32; kk0 += 4) {
      v2f a[2];
      v2f bb[4];
#pragma unroll
      for (int ts = 0; ts < 2; ++ts) {
        int arow = waveT * 32 + ts * 16 + ln;            // M = lane%16
        a[ts] = *(const v2f*)(&lA[arow * 36 + kk0 + k0]); // K0,K0+1 per ISA 16x4 layout
      }
#pragma unroll
      for (int os = 0; os < 4; ++os) {
        int oc = waveO * 64 + os * 16 + ln;              // N = lane%16
        bb[os].x = lB[(kk0 + k0) * 132 + oc];
        bb[os].y = lB[(kk0 + k0 + 1) * 132 + oc];
      }
#pragma unroll
      for (int ts = 0; ts < 2; ++ts)
#pragma unroll
        for (int os = 0; os < 4; ++os)
          acc[ts][os] = __builtin_amdgcn_wmma_f32_16x16x4_f32(
              false, a[ts], false, bb[os], (short)0, acc[ts][os], false, false);
    }
    __syncthreads();
  }

  // Epilogue: D layout (lane%16 = N; VGPR i -> M = i + 8*(lane/16)); fuse bias.
#pragma unroll
  for (int os = 0; os < 4; ++os) {
    int o = oBase + waveO * 64 + os * 16 + ln;
    float bv = bias[o];
#pragma unroll
    for (int ts = 0; ts < 2; ++ts) {
      int t0 = tBase + waveT * 32 + ts * 16 + half * 8;
#pragma unroll
      for (int i = 0; i < 8; ++i) {
        int tt = t0 + i;
        if (tt < TO_)
          Y[((long long)b * TO_ + tt) * OF_ + o] = acc[ts][os][i] + bv;
      }
    }
  }
}

// ---------------------------------------------------------------------------
// Kernel 4: adaptive_avg_pool1d(1023 -> 128) + transpose to [B, 128, 512].
// Deterministic fixed-order reduction (no atomics).
// ---------------------------------------------------------------------------
__global__ __launch_bounds__(512) void pool_kernel(const float* __restrict__ Y,
                                                   float* __restrict__ out) {
  int o = threadIdx.x;            // 0..511
  int w = blockIdx.x & (WIN_ - 1);
  int b = blockIdx.x >> 7;
  int s = (w * TO_) >> 7;                   // floor(w*1023/128)
  int e = ((w + 1) * TO_ + WIN_ - 1) >> 7;  // ceil((w+1)*1023/128)
  float sum = 0.0f;
  for (int t = s; t < e; ++t)
    sum += Y[((long long)b * TO_ + t) * OF_ + o];
  out[((long long)b * WIN_ + w) * OF_ + o] = sum / (float)(e - s);
}

// ---------------------------------------------------------------------------
extern "C" void kernel_launch(void* const* d_in, const int* in_sizes, int n_in,
                              void* d_out, int out_size, void* d_ws, size_t ws_size,
                              hipStream_t stream) {
  const float* x    = (const float*)d_in[0];  // [32,4096,64]
  const float* w    = (const float*)d_in[1];  // [512,576,3]
  const float* bias = (const float*)d_in[2];  // [512]
  float* out = (float*)d_out;                 // [32,128,512]

  // Workspace layout (16B-aligned offsets), total ~146.2 MB:
  //   S  : [32][1025][576] f32 = 75,571,200 B
  //   W2 : [1728][512]     f32 =  3,538,944 B
  //   Y  : [32][1023][512] f32 = 67,043,328 B
  char* ws = (char*)d_ws;
  float* S  = (float*)(ws);
  float* W2 = (float*)(ws + 75571200u);
  float* Y  = (float*)(ws + 75571200u + 3538944u);

  // 32*1025*64 / 256 = 8200 blocks exactly.
  stft_mag_kernel<<<8200, 256, 0, stream>>>(x, S);
  wreorder_kernel<<<(KK_ * OF_ + 255) / 256, 256, 0, stream>>>(w, W2);
  conv_gemm_kernel<<<B_ * 32, 256, 0, stream>>>(S, W2, bias, Y);
  pool_kernel<<<B_ * WIN_, 512, 0, stream>>>(Y, out);
}